// FCOS_20933670601341
// MI455X (gfx1250) — compile-verified
//
#include <hip/hip_runtime.h>
#include <hip/hip_bf16.h>
#include <stdint.h>
#include <stddef.h>

// ---------------- problem constants (must match reference) ----------------
#define NUM_CLASSES 80
#define BATCH       8
#define TOTAL       21824          // sum of level sizes
#define NLEV        5
#define TILE_ROWS   128            // anchors per TDM tile / per block (K1)
#define SCORE_TH    0.2f
#define NMS_TH      0.6f
#define IMG_SZ      1024.0f
#define DETS        100
#define CANDS       3320           // 1000+1000+1000+256+64
#define CAND_PAD    4096           // pow2 pad for bitonic
#define SORT_MAX    16384          // largest level

__constant__ int c_lev_n[NLEV]    = {16384, 4096, 1024, 256, 64};
__constant__ int c_lev_off[NLEV]  = {0, 16384, 20480, 21504, 21760};
__constant__ int c_lev_k[NLEV]    = {1000, 1000, 1000, 256, 64};
__constant__ int c_lev_coff[NLEV] = {0, 1000, 2000, 3000, 3256};

// ---------------- TDM availability ----------------
#if defined(__HIP_DEVICE_COMPILE__) && defined(__gfx1250__) && __has_builtin(__builtin_amdgcn_tensor_load_to_lds)
#define HAVE_TDM 1
typedef unsigned int u32x4 __attribute__((ext_vector_type(4)));
typedef int          i32x8 __attribute__((ext_vector_type(8)));
typedef int          i32x4 __attribute__((ext_vector_type(4)));
#else
#define HAVE_TDM 0
#endif

__device__ __forceinline__ float sigmoidf_(float x) { return 1.0f / (1.0f + expf(-x)); }

// ============================================================================
// Kernel 1: per-anchor class argmax + centerness fusion (memory-bound stage).
// TDM stages a [TILE_ROWS x 80] f32 tile of logits into LDS (OOB rows past the
// tensor end auto-zero). Each wave32 then reduces one row at a time with lanes
// across classes -> bank-conflict-free LDS reads (bank = 16*row + lane), and a
// 5-step shfl_xor butterfly argmax (first-occurrence tie-break, like argmax).
// ============================================================================
__global__ __launch_bounds__(TILE_ROWS)
void fcos_score_kernel(const float* __restrict__ logits,
                       const float* __restrict__ ctrness,
                       float* __restrict__ out_score,
                       int*   __restrict__ out_label) {
  extern __shared__ float k1_tile[];            // TILE_ROWS * NUM_CLASSES floats
  const int b  = blockIdx.y;
  const int a0 = blockIdx.x * TILE_ROWS;
  const float* gsrc = logits + ((size_t)b * TOTAL + a0) * NUM_CLASSES;

#if HAVE_TDM
  if (threadIdx.x < 32) {                       // one TDM issue per block (wave 0)
    const uint64_t ga        = (uint64_t)(uintptr_t)gsrc;
    const uint32_t lds_off   = (uint32_t)(uintptr_t)&k1_tile[0];
    const uint32_t rows_left = (uint32_t)(TOTAL - a0);
    // D# group0: count=1 | lds_addr | global_addr[56:0] | type=2
    u32x4 g0 = { 1u,
                 lds_off,
                 (uint32_t)(ga & 0xFFFFFFFFu),
                 (uint32_t)((ga >> 32) & 0x01FFFFFFu) | 0x80000000u };
    // D# group1: data_size=2 (4B); tensor_dim0=80; tensor_dim1=rows_left;
    //            tile_dim0=80; tile_dim1=TILE_ROWS; tensor_dim0_stride=80
    i32x8 g1 = { (int)(2u << 16),
                 (int)((uint32_t)NUM_CLASSES << 16),            // dim0[15:0] << 16
                 (int)((rows_left & 0xFFFFu) << 16),            // dim0[31:16]=0 | dim1[15:0]
                 (int)(((rows_left >> 16) & 0xFFFFu) |
                       ((uint32_t)NUM_CLASSES << 16)),          // dim1[31:16] | tile_dim0
                 (int)TILE_ROWS,                                // tile_dim1 | tile_dim2=0
                 (int)NUM_CLASSES,                              // dim0_stride[31:0]
                 0, 0 };
    i32x4 gz4 = {0, 0, 0, 0};                                   // 2-D tile: groups 2/3 unused
    i32x8 gz8 = {0, 0, 0, 0, 0, 0, 0, 0};                       // unused trailing group
    __builtin_amdgcn_tensor_load_to_lds(g0, g1, gz4, gz4, gz8, 0);
#if __has_builtin(__builtin_amdgcn_s_wait_tensorcnt)
    __builtin_amdgcn_s_wait_tensorcnt(0);
#else
    asm volatile("s_wait_tensorcnt 0x0" ::: "memory");
#endif
  }
  __syncthreads();
#else
  for (int t = threadIdx.x; t < TILE_ROWS * NUM_CLASSES; t += blockDim.x) {
    const int row = t / NUM_CLASSES;
    k1_tile[t] = (a0 + row < TOTAL) ? gsrc[t] : -1e30f;
  }
  __syncthreads();
#endif

  const int wave  = (int)threadIdx.x >> 5;      // 4 waves/block
  const int lane  = (int)threadIdx.x & 31;
  const int rbase = wave * 32;                  // this wave owns rows [rbase, rbase+32)

  // coalesced centerness preload: lane l caches ctr of row rbase+l
  float myCtr = 0.0f;
  {
    const int a = a0 + rbase + lane;
    if (a < TOTAL) myCtr = ctrness[(size_t)b * TOTAL + a];
  }

  float myScore = -1.0f;
  int   myLabel = 0;

  for (int j = 0; j < 32; ++j) {
    const float* row = &k1_tile[(rbase + j) * NUM_CLASSES];
    // lanes cover classes {lane, lane+32, lane+64} -> conflict-free LDS banks
    float best = row[lane];
    int   bi   = lane;
    const float v1 = row[lane + 32];
    if (v1 > best) { best = v1; bi = lane + 32; }
    if (lane < 16) {
      const float v2 = row[lane + 64];
      if (v2 > best) { best = v2; bi = lane + 64; }
    }
    // butterfly (max, lowest-index) reduction across the wave
#pragma unroll
    for (int off = 16; off >= 1; off >>= 1) {
      const float ov = __shfl_xor(best, off, 32);
      const int   oi = __shfl_xor(bi,   off, 32);
      if (ov > best || (ov == best && oi < bi)) { best = ov; bi = oi; }
    }
    if (lane == j) {                            // row j's result lands in lane j
      const float sc = sqrtf(sigmoidf_(best) * sigmoidf_(myCtr));
      myScore = (sc > SCORE_TH) ? sc : -1.0f;   // masked, as in reference
      myLabel = bi;
    }
  }

  // coalesced stores: lane l writes row rbase+l
  const int a = a0 + rbase + lane;
  if (a < TOTAL) {
    const size_t o = (size_t)b * TOTAL + a;
    out_score[o] = myScore;
    out_label[o] = myLabel;
  }
}

// ============================================================================
// Kernel 2: per-(level,batch) full descending bitonic sort in LDS (level sizes
// are powers of two; 16384 * 8B = 128KB fits the 320KB WGP LDS), then emit the
// top-k decoded & clipped boxes into the padded candidate buffer.
// ============================================================================
__global__ __launch_bounds__(1024)
void fcos_level_topk_kernel(const float* __restrict__ ws_score,
                            const int*   __restrict__ ws_label,
                            const float* __restrict__ reg,
                            const float* __restrict__ anchors,
                            float* __restrict__ cand_score,
                            float* __restrict__ cand_box,
                            int*   __restrict__ cand_label) {
  extern __shared__ char k2_smem[];
  float* s  = (float*)k2_smem;
  int*   id = (int*)(k2_smem + sizeof(float) * SORT_MAX);

  const int lev  = blockIdx.x;
  const int b    = blockIdx.y;
  const int n    = c_lev_n[lev];
  const int off  = c_lev_off[lev];
  const int k    = c_lev_k[lev];
  const int coff = c_lev_coff[lev];
  const int tid  = threadIdx.x;
  const int nthr = blockDim.x;

  const float* src = ws_score + (size_t)b * TOTAL + off;
  for (int j = tid; j < n; j += nthr) { s[j] = src[j]; id[j] = j; }
  __syncthreads();

  // bitonic sort, descending
  for (int size = 2; size <= n; size <<= 1) {
    for (int stride = size >> 1; stride > 0; stride >>= 1) {
      for (int t = tid; t < (n >> 1); t += nthr) {
        const int i = ((t & ~(stride - 1)) << 1) | (t & (stride - 1));
        const int j = i | stride;
        const bool desc = ((i & size) == 0);
        const float si = s[i], sj = s[j];
        if (desc ? (si < sj) : (si > sj)) {
          s[i] = sj; s[j] = si;
          const int ti = id[i]; id[i] = id[j]; id[j] = ti;
        }
      }
      __syncthreads();
    }
  }

  // decode + emit top-k
  for (int j = tid; j < k; j += nthr) {
    const int orig = off + id[j];
    const float* an = anchors + (size_t)orig * 4;
    const float* r  = reg + ((size_t)b * TOTAL + orig) * 4;
    const float cx = (an[0] + an[2]) * 0.5f;
    const float cy = (an[1] + an[3]) * 0.5f;
    const float x0 = fminf(fmaxf(cx - r[0], 0.0f), IMG_SZ);
    const float y0 = fminf(fmaxf(cy - r[1], 0.0f), IMG_SZ);
    const float x1 = fminf(fmaxf(cx + r[2], 0.0f), IMG_SZ);
    const float y1 = fminf(fmaxf(cy + r[3], 0.0f), IMG_SZ);
    const size_t o = (size_t)b * CAND_PAD + coff + j;
    cand_score[o]     = s[j];
    cand_label[o]     = ws_label[(size_t)b * TOTAL + orig];
    cand_box[o * 4 + 0] = x0;
    cand_box[o * 4 + 1] = y0;
    cand_box[o * 4 + 2] = x1;
    cand_box[o * 4 + 3] = y1;
  }
}

// ============================================================================
// Kernel 3: per-batch global sort (4096 padded), greedy NMS with class-offset
// trick, compact first 100 actives, write outputs.
// ============================================================================
__device__ __forceinline__ float iou_off(float4 a, int la, float4 b, int lb) {
  if (la != lb) return 0.0f;                         // offset trick => disjoint
  const float areaA = fmaxf(a.z - a.x, 0.0f) * fmaxf(a.w - a.y, 0.0f);
  const float areaB = fmaxf(b.z - b.x, 0.0f) * fmaxf(b.w - b.y, 0.0f);
  const float ltx = fmaxf(a.x, b.x), lty = fmaxf(a.y, b.y);
  const float rbx = fminf(a.z, b.z), rby = fminf(a.w, b.w);
  const float iw = fmaxf(rbx - ltx, 0.0f), ih = fmaxf(rby - lty, 0.0f);
  const float inter = iw * ih;
  return inter / fmaxf(areaA + areaB - inter, 1e-9f);
}

__global__ __launch_bounds__(1024)
void fcos_nms_kernel(const float* __restrict__ cand_score,
                     const float* __restrict__ cand_box,
                     const int*   __restrict__ cand_label,
                     float* __restrict__ out) {
  extern __shared__ __align__(16) char k3_smem[];
  float4* box = (float4*)k3_smem;                                 // 64 KB
  float*  s   = (float*)(k3_smem + 16 * CAND_PAD);                // 16 KB
  int*    id  = (int*)(k3_smem + 20 * CAND_PAD);                  // 16 KB
  int*    lab = (int*)(k3_smem + 24 * CAND_PAD);                  // 16 KB
  char*   act = (char*)(k3_smem + 28 * CAND_PAD);                 //  4 KB
  __shared__ int sel[DETS];
  __shared__ int s_cnt;

  const int b    = blockIdx.x;
  const int tid  = threadIdx.x;
  const int nthr = blockDim.x;

  for (int j = tid; j < CAND_PAD; j += nthr) {
    s[j]  = (j < CANDS) ? cand_score[(size_t)b * CAND_PAD + j] : -__builtin_inff();
    id[j] = j;
  }
  __syncthreads();

  // bitonic sort (score, idx) descending
  for (int size = 2; size <= CAND_PAD; size <<= 1) {
    for (int stride = size >> 1; stride > 0; stride >>= 1) {
      for (int t = tid; t < (CAND_PAD >> 1); t += nthr) {
        const int i = ((t & ~(stride - 1)) << 1) | (t & (stride - 1));
        const int j = i | stride;
        const bool desc = ((i & size) == 0);
        const float si = s[i], sj = s[j];
        if (desc ? (si < sj) : (si > sj)) {
          s[i] = sj; s[j] = si;
          const int ti = id[i]; id[i] = id[j]; id[j] = ti;
        }
      }
      __syncthreads();
    }
  }

  // gather boxes/labels into sorted order; init active = (score > thresh)
  for (int j = tid; j < CAND_PAD; j += nthr) {
    const size_t o = (size_t)b * CAND_PAD + id[j];
    box[j] = make_float4(cand_box[o * 4 + 0], cand_box[o * 4 + 1],
                         cand_box[o * 4 + 2], cand_box[o * 4 + 3]);
    lab[j] = cand_label[o];
    act[j] = (s[j] > SCORE_TH) ? 1 : 0;
  }
  __syncthreads();

  // greedy NMS: sequential outer (sorted desc), parallel inner
  for (int i = 0; i < CANDS; ++i) {
    if (s[i] <= SCORE_TH) break;                 // uniform: all remaining invalid
    if (act[i]) {
      const float4 bi = box[i];
      const int    li = lab[i];
      for (int j = i + 1 + tid; j < CANDS; j += nthr) {
        if (act[j] && iou_off(bi, li, box[j], lab[j]) > NMS_TH) act[j] = 0;
      }
    }
    __syncthreads();
  }

  // compact first DETS survivors (list already sorted by score)
  if (tid == 0) {
    int cnt = 0;
    for (int j = 0; j < CANDS && cnt < DETS; ++j)
      if (act[j]) sel[cnt++] = j;
    s_cnt = cnt;
  }
  __syncthreads();

  if (tid < DETS) {
    float4 bx = make_float4(0.0f, 0.0f, 0.0f, 0.0f);
    float  sc = 0.0f;
    int    lb = -1;
    if (tid < s_cnt) {
      const int j = sel[tid];
      bx = box[j]; sc = s[j]; lb = lab[j];
    }
    float* ob = out + ((size_t)b * DETS + tid) * 4;      // boxes [8,100,4]
    ob[0] = bx.x; ob[1] = bx.y; ob[2] = bx.z; ob[3] = bx.w;
    out[BATCH * DETS * 4 + b * DETS + tid] = sc;         // scores [8,100]
    ((int*)out)[BATCH * DETS * 5 + b * DETS + tid] = lb; // labels [8,100]
  }
}

// ============================================================================
// launcher
// ============================================================================
extern "C" void kernel_launch(void* const* d_in, const int* in_sizes, int n_in,
                              void* d_out, int out_size, void* d_ws, size_t ws_size,
                              hipStream_t stream) {
  (void)in_sizes; (void)n_in; (void)out_size; (void)ws_size;
  const float* logits  = (const float*)d_in[0];   // (8, 21824, 80)
  const float* reg     = (const float*)d_in[1];   // (8, 21824, 4)
  const float* ctr     = (const float*)d_in[2];   // (8, 21824, 1)
  const float* anchors = (const float*)d_in[3];   // (21824, 4)
  float* out = (float*)d_out;

  char* ws = (char*)d_ws;
  float* ws_score   = (float*)ws; ws += (size_t)BATCH * TOTAL * sizeof(float);
  int*   ws_label   = (int*)ws;   ws += (size_t)BATCH * TOTAL * sizeof(int);
  float* cand_score = (float*)ws; ws += (size_t)BATCH * CAND_PAD * sizeof(float);
  int*   cand_label = (int*)ws;   ws += (size_t)BATCH * CAND_PAD * sizeof(int);
  float* cand_box   = (float*)ws; ws += (size_t)BATCH * CAND_PAD * 4 * sizeof(float);

  dim3 g1((TOTAL + TILE_ROWS - 1) / TILE_ROWS, BATCH);
  fcos_score_kernel<<<g1, TILE_ROWS, TILE_ROWS * NUM_CLASSES * sizeof(float), stream>>>(
      logits, ctr, ws_score, ws_label);

  dim3 g2(NLEV, BATCH);
  fcos_level_topk_kernel<<<g2, 1024, SORT_MAX * (sizeof(float) + sizeof(int)), stream>>>(
      ws_score, ws_label, reg, anchors, cand_score, cand_box, cand_label);

  fcos_nms_kernel<<<BATCH, 1024, CAND_PAD * 29, stream>>>(
      cand_score, cand_box, cand_label, out);
}